// YOLOLayer_58806692217374
// MI455X (gfx1250) — compile-verified
//
#include <hip/hip_runtime.h>
#include <math.h>

// ---------------- constants from the reference ----------------
// B=16, A=3, G=76, NC=80, stride = 608/76 = 8
// scaled anchors = ANCHORS/8: (1.5,2.0) (2.375,4.5) (5.0,3.5)
#define GDIM   76
#define GG     (76*76)             // 5776
#define NCELLS (16*3*GG)           // 277248
#define TILE_CELLS 64
#define NBLK   (NCELLS/TILE_CELLS) // 4332 partial rows
#define NBPAD  4352                // 8 waves * 544, zero-padded rows 4332..4351
#define WAVE_ROWS 544
#define PADROWS (NBPAD - NBLK)     // 20

typedef float v2f __attribute__((ext_vector_type(2)));
typedef float v8f __attribute__((ext_vector_type(8)));

__device__ __forceinline__ float sigm_(float v) { return 1.0f / (1.0f + expf(-v)); }

__device__ __forceinline__ float bce_(float p, float t) {
    float lp = fmaxf(logf(p),        -100.0f);
    float lq = fmaxf(logf(1.0f - p), -100.0f);
    return -(t * lp + (1.0f - t) * lq);
}

// padded pair layout: PP[(row>>1)*32 + k*2 + (row&1)], k = 0..15 (k>=6 zero)
__device__ __forceinline__ int ppIdx(int row, int k) {
    return (row >> 1) * 32 + k * 2 + (row & 1);
}

// ---------------- kernel 1: streaming transform + per-block loss partials ----------------
__global__ __launch_bounds__(256)
void yolo_main(const float* __restrict__ x, const float* __restrict__ iou,
               const unsigned char* __restrict__ objm, const unsigned char* __restrict__ noobjm,
               const float* __restrict__ twi, const float* __restrict__ thi,
               const float* __restrict__ tcls, const float* __restrict__ tconf,
               const float* __restrict__ tbox,
               float* __restrict__ out, float* __restrict__ PP)
{
    __shared__ float tile[85][TILE_CELLS + 1];   // +1 pad: no bank conflicts on transposed read
    __shared__ float rawW[TILE_CELLS];
    __shared__ float rawH[TILE_CELLS];
    __shared__ float red[8][6];

    const int tid = threadIdx.x;
    const int cellBase = blockIdx.x * TILE_CELLS;

    // deterministically zero the WMMA pad rows (ws is never re-poisoned between replays)
    if (tid == 100 && blockIdx.x < PADROWS) {
        int r = NBLK + blockIdx.x;
        #pragma unroll
        for (int k = 0; k < 16; ++k) PP[ppIdx(r, k)] = 0.0f;
    }

    // ---- phase 1: coalesced channel-major reads from x, transform into LDS ----
    for (int j = tid; j < 85 * TILE_CELLS; j += 256) {
        int c  = j >> 6;        // channel 0..84
        int cl = j & 63;        // local cell
        int cell = cellBase + cl;
        int gx = cell % GDIM;
        int t1 = cell / GDIM;
        int gy = t1 % GDIM;
        int t2 = t1 / GDIM;
        int a  = t2 % 3;
        int b  = t2 / 3;
        int xoff = (b * 255 + a * 85 + c) * GG + gy * GDIM + gx;
        if (j + 256 < 85 * TILE_CELLS)
            __builtin_prefetch(x + xoff + 4 * GG, 0, 3);   // gfx1250 global_prefetch_b8
        float v = x[xoff];
        float r;
        if (c == 0) {
            r = (sigm_(v) + (float)gx) * 8.0f;                         // bx*stride
        } else if (c == 1) {
            r = (sigm_(v) + (float)gy) * 8.0f;                         // by*stride
        } else if (c == 2) {
            rawW[cl] = v;
            float aw = (a == 0) ? 1.5f : ((a == 1) ? 2.375f : 5.0f);
            r = expf(v) * aw * 8.0f;                                   // bw*stride
        } else if (c == 3) {
            rawH[cl] = v;
            float ah = (a == 0) ? 2.0f : ((a == 1) ? 4.5f : 3.5f);
            r = expf(v) * ah * 8.0f;                                   // bh*stride
        } else {
            r = sigm_(v);                                              // pconf / pcls
        }
        tile[c][cl] = r;
    }
    __syncthreads();

    // ---- phase 2: fully coalesced channel-minor output writes ----
    {
        const int obase = cellBase * 85;
        for (int j = tid; j < TILE_CELLS * 85; j += 256) {
            int cl = j / 85;
            int c  = j - cl * 85;
            out[obase + j] = tile[c][cl];
        }
    }

    // ---- phase L: loss terms (one lane per cell; obj cells ~2%) ----
    float s0 = 0.f, s1 = 0.f, s2 = 0.f, s3 = 0.f, s4 = 0.f, s5 = 0.f;
    if (tid < TILE_CELLS) {
        int cl = tid;
        int cell = cellBase + cl;
        bool obj   = objm[cell]   != 0;
        bool noobj = noobjm[cell] != 0;
        if (obj || noobj) {
            float pconf = tile[4][cl];
            float bc = bce_(pconf, tconf[cell]);
            if (obj)   { s1 = 1.0f; s3 = bc; }
            if (noobj) { s2 = 1.0f; s4 = bc; }
        }
        if (obj) {
            float bx = tile[0][cl] * 0.125f;    // exact: stride is a power of two
            float by = tile[1][cl] * 0.125f;
            float bw = tile[2][cl] * 0.125f;
            float bh = tile[3][cl] * 0.125f;
            float pw = rawW[cl], ph = rawH[cl];
            float txc = tbox[cell * 4 + 0], tyc = tbox[cell * 4 + 1];
            float twd = tbox[cell * 4 + 2], tht = tbox[cell * 4 + 3];
            float tx1 = txc - twd * 0.5f, ty1 = tyc - tht * 0.5f;
            float tx2 = txc + twd * 0.5f, ty2 = tyc + tht * 0.5f;
            float px1 = bx - bw * 0.5f,  py1 = by - bh * 0.5f;
            float px2 = bx + bw * 0.5f,  py2 = by + bh * 0.5f;
            float xc1 = fminf(px1, tx1), yc1 = fminf(py1, ty1);
            float xc2 = fmaxf(px2, tx2), yc2 = fmaxf(py2, ty2);
            float cdg = (xc2 - xc1) * (xc2 - xc1) + (yc2 - yc1) * (yc2 - yc1) + 1e-7f;
            float dct = (txc - bx) * (txc - bx) + (tyc - by) * (tyc - by);
            float rDIoU = dct / cdg;
            float da = atanf(twi[cell] / thi[cell]) - atanf(pw / ph);
            float vv = 0.40528473456935109f * da * da;   // 4/pi^2
            float iv = iou[cell];
            float S = 1.0f - iv;
            float alpha = vv / (S + vv);
            s0 = 1.0f - iv + rDIoU + alpha * vv;
            float acc = 0.0f;
            #pragma unroll 4
            for (int c = 0; c < 80; ++c) {
                float p = tile[5 + c][cl];
                float t = tcls[cell * 80 + c];
                acc += bce_(p, t);
            }
            s5 = acc;
        }
    }

    // ---- block reduction (wave32 shuffles + LDS across the 8 waves) ----
    float sv[6] = { s0, s1, s2, s3, s4, s5 };
    int lane = tid & 31, wave = tid >> 5;
    #pragma unroll
    for (int k = 0; k < 6; ++k) {
        float v = sv[k];
        for (int m = 16; m > 0; m >>= 1) v += __shfl_xor(v, m, 32);
        if (lane == 0) red[wave][k] = v;
    }
    __syncthreads();
    if (tid == 0) {
        float sums[6];
        #pragma unroll
        for (int k = 0; k < 6; ++k) {
            float v = 0.0f;
            for (int w = 0; w < 8; ++w) v += red[w][k];
            sums[k] = v;
        }
        const int blk = blockIdx.x;
        #pragma unroll
        for (int k = 0; k < 16; ++k)
            PP[ppIdx(blk, k)] = (k < 6) ? sums[k] : 0.0f;
    }
}

// ---------------- kernel 2: totals[6] = ones(1xN) * P(Nx6) via V_WMMA_F32_16X16X4_F32 ----------------
// 8 waves; each wave reduces a 544-row slice of the padded partial matrix with four
// independent WMMA accumulator chains (A = all-ones 16x4 => D[m][n] = column-sum of B).
// Loads are unconditional b64 thanks to the padded pair layout: no exec-mask traffic
// around the WMMAs; EXEC is all ones as the ISA requires.
__global__ __launch_bounds__(256)
void yolo_reduce(const float* __restrict__ PP, float* __restrict__ out, int outIdx)
{
    __shared__ float red2[8][6];
    const int tid  = threadIdx.x;
    const int lane = tid & 31;
    const int wave = tid >> 5;
    const int col  = lane & 15;
    const int half = lane >> 4;

    v8f a0 = { 0.f,0.f,0.f,0.f,0.f,0.f,0.f,0.f };
    v8f a1 = a0, a2 = a0, a3 = a0;
    v2f ones; ones.x = 1.0f; ones.y = 1.0f;

    const int rowBase = wave * WAVE_ROWS;
    for (int c = 0; c < WAVE_ROWS; c += 16) {
        const int pbase = (rowBase + c) >> 1;        // row pair index
        const float* p = PP + col * 2;
        v2f b0 = *(const v2f*)(p + (pbase + 0 + half) * 32);
        v2f b1 = *(const v2f*)(p + (pbase + 2 + half) * 32);
        v2f b2 = *(const v2f*)(p + (pbase + 4 + half) * 32);
        v2f b3 = *(const v2f*)(p + (pbase + 6 + half) * 32);
        a0 = __builtin_amdgcn_wmma_f32_16x16x4_f32(false, ones, false, b0, (short)0, a0, false, false);
        a1 = __builtin_amdgcn_wmma_f32_16x16x4_f32(false, ones, false, b1, (short)0, a1, false, false);
        a2 = __builtin_amdgcn_wmma_f32_16x16x4_f32(false, ones, false, b2, (short)0, a2, false, false);
        a3 = __builtin_amdgcn_wmma_f32_16x16x4_f32(false, ones, false, b3, (short)0, a3, false, false);
    }

    // every D row is identical (A = ones): lane n (n<16) VGPR0 holds the column-n sum
    float t = a0[0] + a1[0] + a2[0] + a3[0];
    if (lane < 6) red2[wave][lane] = t;
    __syncthreads();

    if (tid == 0) {
        float s[6];
        #pragma unroll
        for (int k = 0; k < 6; ++k) {
            float v = 0.0f;
            for (int w = 0; w < 8; ++w) v += red2[w][k];
            s[k] = v;
        }
        float cobj   = fmaxf(s[1], 1.0f);
        float cnoobj = fmaxf(s[2], 1.0f);
        float loss = s[0] / 16.0f                         // CIoU / B
                   + s[3] / cobj + 100.0f * s[4] / cnoobj // conf loss
                   + s[5] / (cobj * 80.0f);               // cls loss
        out[outIdx] = loss;
    }
}

// ---------------- host entry ----------------
extern "C" void kernel_launch(void* const* d_in, const int* in_sizes, int n_in,
                              void* d_out, int out_size, void* d_ws, size_t ws_size,
                              hipStream_t stream)
{
    (void)in_sizes; (void)n_in; (void)ws_size;
    const float*         x      = (const float*)d_in[0];
    const float*         iou    = (const float*)d_in[1];
    const unsigned char* objm   = (const unsigned char*)d_in[2];   // jax bool = 1 byte
    const unsigned char* noobjm = (const unsigned char*)d_in[3];
    const float*         tw     = (const float*)d_in[4];
    const float*         th     = (const float*)d_in[5];
    const float*         tcls   = (const float*)d_in[6];
    const float*         tconf  = (const float*)d_in[7];
    const float*         tbox   = (const float*)d_in[8];
    float* out = (float*)d_out;
    float* PP  = (float*)d_ws;   // (NBPAD/2)*32 floats = 272 KB, fully written every call

    yolo_main<<<NBLK, 256, 0, stream>>>(x, iou, objm, noobjm, tw, th, tcls, tconf, tbox,
                                        out, PP);
    yolo_reduce<<<1, 256, 0, stream>>>(PP, out, out_size - 1);
}